// LNN_12695923327128
// MI455X (gfx1250) — compile-verified
//
#include <hip/hip_runtime.h>
#include <hip/hip_bf16.h>

typedef float v8f __attribute__((ext_vector_type(8)));
typedef float v2f __attribute__((ext_vector_type(2)));

#define NPAR  8
#define DIMX  16
#define HIDN  128
#define BATCH 4096
#define LDSROW 132   // padded row stride (floats): 16B-aligned rows, conflict-free fragment reads

enum { EPI_FWD=0, EPI_BWD=1, EPI_PLAIN=2, EPI_TANF=3, EPI_TANDU4=4, EPI_TANB=5, EPI_TANB1=6 };

__device__ __forceinline__ float softplus_f(float z){
  return (z > 20.0f) ? z : log1pf(__expf(z));
}
__device__ __forceinline__ float sigmoid_f(float z){
  return 1.0f/(1.0f + __expf(-z));
}

__device__ __forceinline__ void epilogue(int mode, float zv, size_t oi, size_t ai,
                                         const float* __restrict__ bias, int col,
                                         const float* __restrict__ aux1,
                                         const float* __restrict__ aux2,
                                         const float* __restrict__ vec, int vstride,
                                         float* __restrict__ out1,
                                         float* __restrict__ out2)
{
  switch(mode){
    case EPI_FWD: {                       // out1=softplus(z+b), out2=sigmoid(z+b)
      float z = zv + bias[col];
      out1[oi] = softplus_f(z);
      out2[ai] = sigmoid_f(z);
    } break;
    case EPI_BWD: {                       // g=acc; out1=u=g*S, out2=e=g*S*(1-S)
      float s = aux1[ai];
      float u = zv * s;
      out1[oi] = u;
      out2[ai] = u * (1.0f - s);
    } break;
    case EPI_PLAIN:  out1[oi] = zv; break;
    case EPI_TANF:                        // dz=acc; out1=dh=dz*S, out2=t=dz*e
      out1[oi] = zv * aux1[ai];
      out2[ai] = zv * aux2[ai];
      break;
    case EPI_TANDU4: out1[oi] = zv * aux1[ai]; break;              // du4 = dz4*e4
    case EPI_TANB:   out1[oi] = zv * aux1[ai] + aux2[ai]; break;   // du = dg*S + t
    case EPI_TANB1:                       // du1 = dg1*S1 + e1*dz1[col]
      out1[oi] = zv * aux1[ai] + aux2[ai] * vec[(size_t)col * vstride];
      break;
  }
}

// ---------------------------------------------------------------------------
// Big GEMM: OUT[BATCH x 128] = epi( A[BATCH x 128] * op(B)[128 x 128] )
// One 256-thread block per 16-row M-tile; A tile staged once into LDS via the
// CDNA5 async Global->LDS path; wave w computes N-tile w with
// V_WMMA_F32_16X16X4_F32 (fp32 end-to-end).
//   BTRANS=1 : op(B)[k][n] = B[n*ldb + k]   (A @ W^T)
//   BTRANS=0 : op(B)[k][n] = B[k*ldb + n]   (A @ W)
// ---------------------------------------------------------------------------
template<int BTRANS>
__global__ __launch_bounds__(256) void gemm128(
    const float* __restrict__ A,
    const float* __restrict__ B, int ldb,
    const float* __restrict__ bias,
    const float* __restrict__ aux1,
    const float* __restrict__ aux2,
    const float* __restrict__ vec, int vstride,
    float* __restrict__ out1,
    float* __restrict__ out2,
    int mode)
{
  __shared__ __align__(16) float As[16 * LDSROW];   // 8448 B
  const int tid  = threadIdx.x;
  const int lane = tid & 31;
  const int wv   = tid >> 5;            // n-tile 0..7
  const int mt   = blockIdx.x;          // m-tile 0..255

  // ---- stage A tile (16 x 128 f32) via async copy, 16B chunks ----
  const float* Abase = A + (size_t)mt * 16 * HIDN;
  #pragma unroll
  for (int i = 0; i < 2; ++i){
    int chunk = tid + i*256;            // 0..511 (512 x float4)
    int row = chunk >> 5;               // 32 chunks per row
    int c4  = chunk & 31;
    unsigned ldsoff = (unsigned)(size_t)&As[row*LDSROW + c4*4];
    unsigned long long ga = (unsigned long long)(size_t)(Abase + row*HIDN + c4*4);
    asm volatile("global_load_async_to_lds_b128 %0, %1, off"
                 :: "v"(ldsoff), "v"(ga) : "memory");
  }
  asm volatile("s_wait_asynccnt 0x0" ::: "memory");
  __syncthreads();

  const int lm = lane & 15;
  const int kh = lane >> 4;             // 0/1
  const int bcol = wv*16 + lm;

  v8f acc = {0.f,0.f,0.f,0.f,0.f,0.f,0.f,0.f};
  #pragma unroll 8
  for (int k = 0; k < HIDN; k += 4){
    const int ka = k + 2*kh;            // fp32 16x4 A layout: K = kbase + 2*(lane/16) + {0,1}
    v2f a = *(const v2f*)&As[lm*LDSROW + ka];
    v2f b;
    if (BTRANS){
      b = *(const v2f*)(B + (size_t)bcol*ldb + ka);
    } else {
      b.x = B[(size_t)ka    *ldb + bcol];
      b.y = B[(size_t)(ka+1)*ldb + bcol];
    }
    acc = __builtin_amdgcn_wmma_f32_16x16x4_f32(false, a, false, b, (short)0, acc,
                                                false, false);
  }

  const int col   = wv*16 + lm;
  const int rbase = mt*16 + 8*kh;
  #pragma unroll
  for (int vv = 0; vv < 8; ++vv){
    const int r = rbase + vv;
    const size_t oi = (size_t)r*HIDN + col;
    epilogue(mode, acc[vv], oi, oi, bias, col, aux1, aux2, vec, vstride, out1, out2);
  }
}

// ---------------------------------------------------------------------------
// Generic small GEMM (direct loads): used for K=16 (layer 1) and N=16 (J, Hcols)
// ---------------------------------------------------------------------------
template<int BTRANS>
__global__ __launch_bounds__(256) void gemm_small(
    const float* __restrict__ A, int lda,
    const float* __restrict__ B, int ldb,
    const float* __restrict__ bias,
    float* __restrict__ out1, int ldo,
    float* __restrict__ out2,
    int M, int N, int K, int mode)
{
  const int lane = threadIdx.x & 31;
  const int wave = (int)((blockIdx.x * blockDim.x + threadIdx.x) >> 5);
  const int tiles_n = N >> 4;
  const int mt = wave / tiles_n;
  const int nt = wave - mt * tiles_n;

  const int lm = lane & 15;
  const int kh = lane >> 4;
  const int arow = mt*16 + lm;
  const int bcol = nt*16 + lm;

  v8f acc = {0.f,0.f,0.f,0.f,0.f,0.f,0.f,0.f};
  for (int k = 0; k < K; k += 4){
    const int ka = k + 2*kh;
    v2f a = *(const v2f*)(A + (size_t)arow*lda + ka);
    v2f b;
    if (BTRANS){
      b = *(const v2f*)(B + (size_t)bcol*ldb + ka);
    } else {
      b.x = B[(size_t)ka    *ldb + bcol];
      b.y = B[(size_t)(ka+1)*ldb + bcol];
    }
    acc = __builtin_amdgcn_wmma_f32_16x16x4_f32(false, a, false, b, (short)0, acc,
                                                false, false);
  }

  const int col   = nt*16 + lm;
  const int rbase = mt*16 + 8*kh;
  #pragma unroll
  for (int vv = 0; vv < 8; ++vv){
    const int r = rbase + vv;
    epilogue(mode, acc[vv], (size_t)r*ldo + col, (size_t)r*HIDN + col,
             bias, col, nullptr, nullptr, nullptr, 0, out1, out2);
  }
}

// u4 = W5 (*) S4 ; e4 = W5 (*) S4 (*) (1-S4)
__global__ __launch_bounds__(256) void ew_u4(const float* __restrict__ S4,
                                             const float* __restrict__ W5,
                                             float* __restrict__ u4,
                                             float* __restrict__ e4, int total)
{
  int i = blockIdx.x * blockDim.x + threadIdx.x;
  if (i >= total) return;
  int n = i & (HIDN-1);
  float s = S4[i], w = W5[n];
  u4[i] = w * s;
  e4[i] = w * s * (1.0f - s);
}

// dh1 = S1 (*) broadcast-column(W1[:,8+j])
__global__ __launch_bounds__(256) void ew_scale(const float* __restrict__ S,
                                                const float* __restrict__ colv, int vstride,
                                                float* __restrict__ outp, int total)
{
  int i = blockIdx.x * blockDim.x + threadIdx.x;
  if (i >= total) return;
  outp[i] = S[i] * colv[(size_t)(i & (HIDN-1)) * vstride];
}

// Per-sample: B = H[8:,8:], C = H[8:,:8]; rhs = J[:8] - C @ x[8:];
// pivoted Gauss-Jordan with tiny-pivot guard (pinv-ish).
__global__ __launch_bounds__(128) void solve8(const float* __restrict__ Hb,   // [8][BATCH][16]
                                              const float* __restrict__ Jb,   // [BATCH][16]
                                              const float* __restrict__ X,    // [BATCH][16]
                                              float* __restrict__ out, int M)
{
  int m = blockIdx.x * blockDim.x + threadIdx.x;
  if (m >= M) return;
  float xq[NPAR];
  #pragma unroll
  for (int i = 0; i < NPAR; ++i) xq[i] = X[(size_t)m*DIMX + NPAR + i];

  float Bm[NPAR][NPAR], rhs[NPAR];
  #pragma unroll
  for (int j = 0; j < NPAR; ++j){
    const float* hc = Hb + ((size_t)j*M + m)*DIMX;   // Hcol_j[i] = H[i, 8+j]
    float dot = 0.f;
    #pragma unroll
    for (int i = 0; i < NPAR; ++i){
      dot += hc[i] * xq[i];                          // C[j][i] = Hcol_j[i]
      Bm[i][j] = hc[NPAR + i];                       // B[i][j] = Hcol_j[8+i]
    }
    rhs[j] = Jb[(size_t)m*DIMX + j] - dot;
  }
  for (int a = 0; a < NPAR; ++a)
    for (int b = a+1; b < NPAR; ++b){
      float t = 0.5f*(Bm[a][b] + Bm[b][a]); Bm[a][b] = t; Bm[b][a] = t;
    }
  for (int c = 0; c < NPAR; ++c){
    int p = c; float mx = fabsf(Bm[c][c]);
    for (int r = c+1; r < NPAR; ++r){ float a = fabsf(Bm[r][c]); if (a > mx){ mx = a; p = r; } }
    if (p != c){
      for (int cc = 0; cc < NPAR; ++cc){ float t = Bm[c][cc]; Bm[c][cc] = Bm[p][cc]; Bm[p][cc] = t; }
      float t = rhs[c]; rhs[c] = rhs[p]; rhs[p] = t;
    }
    float piv = Bm[c][c];
    float inv = (fabsf(piv) > 1e-8f) ? 1.0f/piv : 0.0f;
    for (int cc = c; cc < NPAR; ++cc) Bm[c][cc] *= inv;
    rhs[c] *= inv;
    for (int r = 0; r < NPAR; ++r){
      if (r == c) continue;
      float f = Bm[r][c];
      for (int cc = c; cc < NPAR; ++cc) Bm[r][cc] -= f * Bm[c][cc];
      rhs[r] -= f * rhs[c];
    }
  }
  #pragma unroll
  for (int j = 0; j < NPAR; ++j) out[(size_t)m*NPAR + j] = rhs[j];
}

extern "C" void kernel_launch(void* const* d_in, const int* in_sizes, int n_in,
                              void* d_out, int out_size, void* d_ws, size_t ws_size,
                              hipStream_t stream)
{
  const float* X  = (const float*)d_in[0];
  const float* W1 = (const float*)d_in[1];
  const float* b1 = (const float*)d_in[2];
  const float* W2 = (const float*)d_in[3];
  const float* b2 = (const float*)d_in[4];
  const float* W3 = (const float*)d_in[5];
  const float* b3 = (const float*)d_in[6];
  const float* W4 = (const float*)d_in[7];
  const float* b4 = (const float*)d_in[8];
  const float* W5 = (const float*)d_in[9];
  // b5 vanishes under grad/hessian
  float* out = (float*)d_out;

  float* ws = (float*)d_ws;
  const size_t per = (size_t)BATCH * HIDN;          // 2 MB each; ~30 MB total
  float *S1 = ws + 0*per, *S2 = ws + 1*per, *S3 = ws + 2*per, *S4 = ws + 3*per;
  float *E1 = ws + 4*per, *E2 = ws + 5*per, *E3 = ws + 6*per, *E4 = ws + 7*per;
  float *U1 = ws + 8*per;
  float *T0 = ws + 9*per, *T1 = ws + 10*per, *T2 = ws + 11*per, *T3 = ws + 12*per;
  float *Jb = ws + 13*per;                          // [BATCH][16]
  float *Hb = Jb + (size_t)BATCH*DIMX;              // [8][BATCH][16]

  const dim3 blk(256), grdBig(BATCH/16);            // 256 blocks x 8 waves

  auto gb = [&](bool btr, int mode, const float* A, const float* B, const float* bias,
                const float* a1, const float* a2, const float* vec, int vstr,
                float* o1, float* o2){
    if (btr) gemm128<1><<<grdBig, blk, 0, stream>>>(A, B, HIDN, bias, a1, a2, vec, vstr, o1, o2, mode);
    else     gemm128<0><<<grdBig, blk, 0, stream>>>(A, B, HIDN, bias, a1, a2, vec, vstr, o1, o2, mode);
  };
  auto gs = [&](bool btr, int mode, const float* A, int lda, const float* B, int ldb,
                const float* bias, float* o1, int ldo, float* o2, int N, int K){
    dim3 grd(((BATCH/16)*(N/16))/8);
    if (btr) gemm_small<1><<<grd, blk, 0, stream>>>(A, lda, B, ldb, bias, o1, ldo, o2, BATCH, N, K, mode);
    else     gemm_small<0><<<grd, blk, 0, stream>>>(A, lda, B, ldb, bias, o1, ldo, o2, BATCH, N, K, mode);
  };

  // ---- forward: H_l -> T*, S_l ----
  gs(true, EPI_FWD, X, DIMX, W1, DIMX, b1, T0, HIDN, S1, HIDN, DIMX);     // layer 1 (K=16)
  gb(true, EPI_FWD, T0, W2, b2, nullptr, nullptr, nullptr, 0, T1, S2);
  gb(true, EPI_FWD, T1, W3, b3, nullptr, nullptr, nullptr, 0, T2, S3);
  gb(true, EPI_FWD, T2, W4, b4, nullptr, nullptr, nullptr, 0, T3, S4);    // T3 = trash

  // ---- reverse: u_l, e_l, J ----
  ew_u4<<<(BATCH*HIDN)/256, 256, 0, stream>>>(S4, W5, T0, E4, BATCH*HIDN);              // u4 -> T0
  gb(false, EPI_BWD, T0, W4, nullptr, S3, nullptr, nullptr, 0, T1, E3);                 // u3 -> T1
  gb(false, EPI_BWD, T1, W3, nullptr, S2, nullptr, nullptr, 0, T2, E2);                 // u2 -> T2
  gb(false, EPI_BWD, T2, W2, nullptr, S1, nullptr, nullptr, 0, U1, E1);                 // u1
  gs(false, EPI_PLAIN, U1, HIDN, W1, DIMX, nullptr, Jb, DIMX, nullptr, DIMX, HIDN);     // J

  // ---- 8 Hessian-vector products (tangent e_{8+j}; sample-independent) ----
  for (int j = 0; j < NPAR; ++j){
    const float* c1 = W1 + NPAR + j;   // dz1[k] = W1[k][8+j], stride DIMX
    ew_scale<<<(BATCH*HIDN)/256, 256, 0, stream>>>(S1, c1, DIMX, T2, BATCH*HIDN);       // dh1 -> T2
    gb(true,  EPI_TANF,   T2, W2, nullptr, S2, E2, nullptr, 0, T0, T1);  // dh2->T0, t2->T1
    gb(true,  EPI_TANF,   T0, W3, nullptr, S3, E3, nullptr, 0, T2, T3);  // dh3->T2, t3->T3
    gb(true,  EPI_TANDU4, T2, W4, nullptr, E4, nullptr, nullptr, 0, T0, nullptr);       // du4->T0
    gb(false, EPI_TANB,   T0, W4, nullptr, S3, T3, nullptr, 0, T2, nullptr);            // du3->T2
    gb(false, EPI_TANB,   T2, W3, nullptr, S2, T1, nullptr, 0, T0, nullptr);            // du2->T0
    gb(false, EPI_TANB1,  T0, W2, nullptr, S1, E1, c1, DIMX, T2, nullptr);              // du1->T2
    gs(false, EPI_PLAIN,  T2, HIDN, W1, DIMX, nullptr,
       Hb + (size_t)j*BATCH*DIMX, DIMX, nullptr, DIMX, HIDN);                           // Hcol_j
  }

  // ---- per-sample 8x8 guarded solve ----
  solve8<<<BATCH/128, 128, 0, stream>>>(Hb, Jb, X, out, BATCH);
}